// HierarchicalDense_61727269978567
// MI455X (gfx1250) — compile-verified
//
#include <hip/hip_runtime.h>

typedef __attribute__((ext_vector_type(2))) float v2f;
typedef __attribute__((ext_vector_type(8))) float v8f;

#define B_DIM 16384
#define C_DIM 128
#define F_DIM 5000

// ---------------------------------------------------------------------------
// Kernel 1: probs = sigmoid(x @ W + b), written to out[B,F] (row-major).
// Block = 256 threads = 8 wave32s arranged 2(M) x 4(N); block tile 32M x 64N.
// Each wave computes one 16x16 tile via 32 chained V_WMMA_F32_16X16X4_F32.
// A tile staged in LDS (row stride 132 floats to avoid bank conflicts).
// ---------------------------------------------------------------------------
__global__ __launch_bounds__(256)
void gemm_sigmoid_kernel(const float* __restrict__ x,
                         const float* __restrict__ W,
                         const float* __restrict__ bias,
                         float* __restrict__ out)
{
    __shared__ float As[32 * 132];   // 32 rows x 128 K, padded

    const int tid  = threadIdx.x;
    const int lane = tid & 31;
    const int wave = tid >> 5;
    const int wrow = wave >> 2;      // 0..1 : M sub-tile
    const int wcol = wave & 3;       // 0..3 : N sub-tile
    const int half = lane >> 4;      // 0..1 : lane half (K split per WMMA layout)
    const int l16  = lane & 15;

    const int m_base = blockIdx.y * 32;          // always in-range (16384 % 32 == 0)
    const int n_base = blockIdx.x * 64;

    // Stage A tile (32x128 f32) into LDS, fully coalesced float4 loads.
    {
        const float* src = x + (size_t)m_base * C_DIM;
        #pragma unroll
        for (int chunk = 0; chunk < 4; ++chunk) {
            int flat = (chunk * 256 + tid) * 4;  // 0..4095, step 4
            int r = flat >> 7;                    // /128
            int c = flat & 127;
            float4 v = *(const float4*)(src + r * C_DIM + c);
            *(float4*)(&As[r * 132 + c]) = v;
        }
    }
    __syncthreads();

    const int m_local = wrow * 16 + l16;
    const int ncol    = n_base + wcol * 16 + l16;
    const int nc      = (ncol < F_DIM) ? ncol : (F_DIM - 1);   // clamp loads, no divergence

    v8f acc = {};
    #pragma unroll
    for (int k0 = 0; k0 < C_DIM; k0 += 4) {
        // A 16x4: lanes0-15 v={K0,K1}, lanes16-31 v={K2,K3} -> contiguous pair in LDS
        v2f a = *(const v2f*)(&As[m_local * 132 + k0 + half * 2]);
        // B 4x16: v0 = W[k0+2h][n], v1 = W[k0+2h+1][n]
        v2f b;
        b.x = W[(k0 + 2 * half)     * F_DIM + nc];
        b.y = W[(k0 + 2 * half + 1) * F_DIM + nc];
        acc = __builtin_amdgcn_wmma_f32_16x16x4_f32(
                /*neg_a=*/false, a, /*neg_b=*/false, b,
                /*c_mod=*/(short)0, acc, /*reuse_a=*/false, /*reuse_b=*/false);
    }

    const float bb = bias[nc];
    #pragma unroll
    for (int r = 0; r < 8; ++r) {
        // C/D layout: VGPR r -> row r (lanes 0-15) / row r+8 (lanes 16-31)
        float t = acc[r] + bb;
        float s = 1.0f / (1.0f + __expf(-t));
        int mg = m_base + wrow * 16 + r + half * 8;
        if (ncol < F_DIM)
            out[(size_t)mg * F_DIM + ncol] = s;
    }
}

// ---------------------------------------------------------------------------
// Kernel 2: in-place hierarchical segment-max per batch row.
// Block b owns row b of out exclusively: gather probs[row][child], LDS
// atomic-max into smax[parent] (all sigmoid values > 0 => unsigned-bit max
// == float max, and 0-init is safe because every parent has a self-edge),
// then overwrite the row. No global atomics, no workspace.
// ---------------------------------------------------------------------------
__global__ __launch_bounds__(256)
void hier_max_kernel(float* __restrict__ probs,
                     const int* __restrict__ edge_parent,
                     const int* __restrict__ edge_child,
                     int E)
{
    __shared__ unsigned smax[F_DIM];   // 20000 bytes
    const int tid = threadIdx.x;
    float* row = probs + (size_t)blockIdx.x * F_DIM;

    for (int j = tid; j < F_DIM; j += 256) smax[j] = 0u;
    __syncthreads();

    for (int e = tid; e < E; e += 256) {
        int c = edge_child[e];
        int p = edge_parent[e];
        float v = row[c];
        atomicMax(&smax[p], __float_as_uint(v));   // ds_max_u32
    }
    __syncthreads();

    for (int j = tid; j < F_DIM; j += 256)
        row[j] = __uint_as_float(smax[j]);
}

extern "C" void kernel_launch(void* const* d_in, const int* in_sizes, int n_in,
                              void* d_out, int out_size, void* d_ws, size_t ws_size,
                              hipStream_t stream)
{
    const float* x    = (const float*)d_in[0];   // [B, C]
    const float* W    = (const float*)d_in[1];   // [C, F]
    const float* bias = (const float*)d_in[2];   // [F]
    const int* ep     = (const int*)d_in[3];     // [E]
    const int* ec     = (const int*)d_in[4];     // [E]
    const int E       = in_sizes[3];
    float* out        = (float*)d_out;           // [B, F]

    dim3 grid1((F_DIM + 63) / 64, B_DIM / 32);   // 79 x 512
    gemm_sigmoid_kernel<<<grid1, 256, 0, stream>>>(x, W, bias, out);

    hier_max_kernel<<<B_DIM, 256, 0, stream>>>(out, ep, ec, E);
}